// ContentSelector_75445395522264
// MI455X (gfx1250) — compile-verified
//
#include <hip/hip_runtime.h>
#include <hip/hip_bf16.h>
#include <math.h>

// ---- problem constants ----
#define Dm   1024
#define Bsz  8
#define NS   4096
#define NE   1024

// ---- workspace layout (float offsets) ----
#define WS_H      0        // [8192]  h
#define WS_C      8192     // [8192]  c
#define WS_ECTX   16384    // [8192]  entity context
#define WS_SCTX   24576    // [8192]  sentence context
#define WS_SCORES 32768    // [32768] attention scores (max N=4096)
#define WS_RMAX   65536    // [8]     per-batch score max
#define WS_PSUM   66560    // [256]   partial sum-of-exp (B x nch)
#define WS_TOP    67584    // [1]     (int) selected sentence index
#define WS_XPAD   68608    // [16*2048] padded LSTM input A-matrix
#define WS_HPAD   101376   // [16*1024] padded h A-matrix
#define WS_GATES  117760   // [8*4096] LSTM gates
#define WS_PART   150528   // [8*32*1024] partial weighted sums

typedef __attribute__((ext_vector_type(2))) float v2f;
typedef __attribute__((ext_vector_type(8))) float v8f;

// Zero h, c, and the padded A-matrices (rows 8..15 must stay zero for WMMA).
__global__ void init_kernel(float* ws) {
    int i = blockIdx.x * 256 + threadIdx.x;   // 65536 threads
    if (i < 16384)       ws[WS_H + i] = 0.0f;            // h and c (contiguous)
    else if (i < 49152)  ws[WS_XPAD + (i - 16384)] = 0.0f;
    else                 ws[WS_HPAD + (i - 49152)] = 0.0f;
}

// scores[b,n] = dot(mem[b,n,:], W[D:]) + dot(h[b,:], W[:D]) + bias
__global__ void scores_kernel(const float* __restrict__ mem,
                              const float* __restrict__ Wfull,
                              const float* __restrict__ bias,
                              const float* __restrict__ h,
                              int N, float* __restrict__ scores) {
    __shared__ float sW2[Dm];
    __shared__ float red[256];
    const int b = blockIdx.y;
    const int tid = threadIdx.x;
    for (int i = tid; i < Dm; i += 256) sW2[i] = Wfull[Dm + i];
    // block-wide hdot
    float hp = 0.0f;
    for (int i = tid; i < Dm; i += 256) hp += h[b * Dm + i] * Wfull[i];
    red[tid] = hp; __syncthreads();
    for (int s = 128; s > 0; s >>= 1) {
        if (tid < s) red[tid] += red[tid + s];
        __syncthreads();
    }
    const float hdot = red[0];
    const float bv = bias[0];

    const int wave = tid >> 5, lane = tid & 31;
    const int row0 = blockIdx.x * 64 + wave * 8;   // 8 waves x 8 rows = 64 rows/block
    const float4* wp = (const float4*)sW2;
    for (int r = 0; r < 8; ++r) {
        const int n = row0 + r;
        const float4* mp = (const float4*)(mem + ((size_t)b * N + n) * Dm);
        float acc = 0.0f;
        #pragma unroll
        for (int q = 0; q < 8; ++q) {
            float4 mv = mp[lane + 32 * q];
            float4 wv = wp[lane + 32 * q];
            acc += mv.x * wv.x + mv.y * wv.y + mv.z * wv.z + mv.w * wv.w;
        }
        #pragma unroll
        for (int off = 16; off > 0; off >>= 1) acc += __shfl_down(acc, off, 32);
        if (lane == 0) scores[(size_t)b * N + n] = acc + hdot + bv;
    }
}

__global__ void rowmax_kernel(const float* __restrict__ scores, int N,
                              float* __restrict__ rmax) {
    const int b = blockIdx.x, tid = threadIdx.x;
    __shared__ float red[256];
    float m = -INFINITY;
    for (int n = tid; n < N; n += 256) m = fmaxf(m, scores[(size_t)b * N + n]);
    red[tid] = m; __syncthreads();
    for (int s = 128; s > 0; s >>= 1) {
        if (tid < s) red[tid] = fmaxf(red[tid], red[tid + s]);
        __syncthreads();
    }
    if (tid == 0) rmax[b] = red[0];
}

// Per (b, chunk of 128 rows): e=exp(s-max); psum += e; part[d] += e*mem[n,d]
__global__ void attend_part_kernel(const float* __restrict__ mem,
                                   const float* __restrict__ scores,
                                   const float* __restrict__ rmax,
                                   int N, int nch,
                                   float* __restrict__ part,
                                   float* __restrict__ psum) {
    const int b = blockIdx.y, ch = blockIdx.x, tid = threadIdx.x;
    const int row0 = ch * 128;
    __shared__ float e[128];
    __shared__ float red[256];
    const float m = rmax[b];
    float ep = 0.0f;
    if (tid < 128) {
        float v = expf(scores[(size_t)b * N + row0 + tid] - m);
        e[tid] = v;
        ep = v;
    }
    red[tid] = ep; __syncthreads();
    for (int s = 128; s > 0; s >>= 1) {
        if (tid < s) red[tid] += red[tid + s];
        __syncthreads();
    }
    if (tid == 0) psum[b * nch + ch] = red[0];

    float4 acc = {0.0f, 0.0f, 0.0f, 0.0f};
    const float4* base = (const float4*)(mem + ((size_t)b * N + row0) * Dm);
    for (int r = 0; r < 128; ++r) {
        __builtin_prefetch(&base[(r + 2) * 256 + tid], 0, 0);
        float4 mv = base[r * 256 + tid];
        float w = e[r];
        acc.x += w * mv.x; acc.y += w * mv.y;
        acc.z += w * mv.z; acc.w += w * mv.w;
    }
    ((float4*)(part + (size_t)(b * nch + ch) * Dm))[tid] = acc;
}

__global__ void attend_reduce_kernel(const float* __restrict__ part,
                                     const float* __restrict__ psum,
                                     int nch, float* __restrict__ ctx) {
    const int b = blockIdx.y;
    const int d = blockIdx.x * 256 + threadIdx.x;
    float se = 0.0f, s = 0.0f;
    for (int ch = 0; ch < nch; ++ch) {
        se += psum[b * nch + ch];
        s += part[(size_t)(b * nch + ch) * Dm + d];
    }
    ctx[b * Dm + d] = s / se;
}

// p[b] = sigmoid([h,ectx,sctx] @ Wp + bp); top = argmax_b p (first max wins)
__global__ void p_select_kernel(const float* __restrict__ h,
                                const float* __restrict__ ectx,
                                const float* __restrict__ sctx,
                                const float* __restrict__ Wp,
                                const float* __restrict__ bp,
                                int* __restrict__ top) {
    const int tid = threadIdx.x;
    const int b = tid >> 5, lane = tid & 31;
    __shared__ float sp[8];
    float acc = 0.0f;
    for (int d = lane; d < Dm; d += 32) {
        acc += h[b * Dm + d] * Wp[d]
             + ectx[b * Dm + d] * Wp[Dm + d]
             + sctx[b * Dm + d] * Wp[2 * Dm + d];
    }
    #pragma unroll
    for (int off = 16; off > 0; off >>= 1) acc += __shfl_down(acc, off, 32);
    if (lane == 0) sp[b] = 1.0f / (1.0f + expf(-(acc + bp[0])));
    __syncthreads();
    if (tid == 0) {
        float best = sp[0]; int bi = 0;
        for (int i = 1; i < Bsz; ++i) if (sp[i] > best) { best = sp[i]; bi = i; }
        top[0] = bi;
    }
}

// Build xpad=[sel_sent, ectx] (rows 0..7), hpad=h, and write the output slot.
__global__ void gather_kernel(const float* __restrict__ sents,
                              const float* __restrict__ ectx,
                              const float* __restrict__ h,
                              const int* __restrict__ top, int t,
                              float* __restrict__ xpad,
                              float* __restrict__ hpad,
                              float* __restrict__ out) {
    const int idx = blockIdx.x * 256 + threadIdx.x;  // B*2048 = 16384
    const int b = idx >> 11, d = idx & 2047;
    const int ti = top[0];
    if (d < Dm) {
        float v = sents[((size_t)b * NS + ti) * Dm + d];
        xpad[b * 2048 + d] = v;
        out[((size_t)b * 3 + t) * Dm + d] = v;
        hpad[b * Dm + d] = h[b * Dm + d];
    } else {
        xpad[b * 2048 + d] = ectx[b * Dm + (d - Dm)];
    }
}

// gates[16x4096] = Xpad[16x2048] @ W_ih^T + Hpad[16x1024] @ W_hh^T + biases
// via V_WMMA_F32_16X16X4_F32; one wave per 16-wide N tile, K-loop of 4.
__global__ void gates_wmma_kernel(const float* __restrict__ xpad,
                                  const float* __restrict__ hpad,
                                  const float* __restrict__ W_ih,
                                  const float* __restrict__ W_hh,
                                  const float* __restrict__ b_ih,
                                  const float* __restrict__ b_hh,
                                  float* __restrict__ gates) {
    const int wave = threadIdx.x >> 5;
    const int lane = threadIdx.x & 31;
    const int tile = blockIdx.x * 4 + wave;     // 64 blocks x 4 waves = 256 tiles
    const int n0 = tile * 16;
    const int m = lane & 15;                    // A-matrix row (M)
    const int hk = (lane >> 4) << 1;            // K sub-offset: 0 or 2
    const int j = n0 + m;                       // weight row served by this lane (B op)

    v8f acc = {};
    {
        const float* xr = xpad + m * 2048 + hk;
        const float* wr = W_ih + (size_t)j * 2048 + hk;
        #pragma unroll 4
        for (int k = 0; k < 2048; k += 4) {
            v2f a = *(const v2f*)(xr + k);
            v2f bb = *(const v2f*)(wr + k);
            acc = __builtin_amdgcn_wmma_f32_16x16x4_f32(
                false, a, false, bb, (short)0, acc, false, false);
        }
    }
    {
        const float* hr = hpad + m * 1024 + hk;
        const float* wr = W_hh + (size_t)j * 1024 + hk;
        #pragma unroll 4
        for (int k = 0; k < 1024; k += 4) {
            v2f a = *(const v2f*)(hr + k);
            v2f bb = *(const v2f*)(wr + k);
            acc = __builtin_amdgcn_wmma_f32_16x16x4_f32(
                false, a, false, bb, (short)0, acc, false, false);
        }
    }
    // D layout: VGPR v holds M=v (lanes 0-15), M=v+8 (lanes 16-31). Batch rows 0..7.
    if (lane < 16) {
        const int col = n0 + lane;
        const float bias = b_ih[col] + b_hh[col];
        #pragma unroll
        for (int v = 0; v < 8; ++v) gates[v * 4096 + col] = acc[v] + bias;
    }
}

// c = sig(f)*c + sig(i)*tanh(g); h = sig(o)*tanh(c)   (gate order i,f,g,o)
__global__ void lstm_kernel(const float* __restrict__ gates,
                            float* __restrict__ h, float* __restrict__ c) {
    const int idx = blockIdx.x * 256 + threadIdx.x;  // 8192
    const int b = idx >> 10, d = idx & 1023;
    const float* g = gates + b * 4096;
    float gi = g[d], gf = g[1024 + d], gg = g[2048 + d], go = g[3072 + d];
    float si = 1.0f / (1.0f + expf(-gi));
    float sf = 1.0f / (1.0f + expf(-gf));
    float so = 1.0f / (1.0f + expf(-go));
    float cn = sf * c[idx] + si * tanhf(gg);
    c[idx] = cn;
    h[idx] = so * tanhf(cn);
}

extern "C" void kernel_launch(void* const* d_in, const int* in_sizes, int n_in,
                              void* d_out, int out_size, void* d_ws, size_t ws_size,
                              hipStream_t stream) {
    const float* sents    = (const float*)d_in[0];
    const float* entities = (const float*)d_in[1];
    const float* Wae      = (const float*)d_in[2];
    const float* bae      = (const float*)d_in[3];
    const float* Was      = (const float*)d_in[4];
    const float* bas      = (const float*)d_in[5];
    const float* Wp       = (const float*)d_in[6];
    const float* bp       = (const float*)d_in[7];
    const float* W_ih     = (const float*)d_in[8];
    const float* W_hh     = (const float*)d_in[9];
    const float* b_ih     = (const float*)d_in[10];
    const float* b_hh     = (const float*)d_in[11];
    float* out = (float*)d_out;

    float* ws = (float*)d_ws;
    float* f_h    = ws + WS_H;
    float* f_c    = ws + WS_C;
    float* f_ectx = ws + WS_ECTX;
    float* f_sctx = ws + WS_SCTX;
    float* f_sc   = ws + WS_SCORES;
    float* f_rmax = ws + WS_RMAX;
    float* f_psum = ws + WS_PSUM;
    int*   i_top  = (int*)(ws + WS_TOP);
    float* f_xpad = ws + WS_XPAD;
    float* f_hpad = ws + WS_HPAD;
    float* f_gate = ws + WS_GATES;
    float* f_part = ws + WS_PART;

    init_kernel<<<256, 256, 0, stream>>>(ws);

    for (int t = 0; t < 3; ++t) {
        // ---- attend over entities -> ectx ----
        scores_kernel<<<dim3(NE / 64, Bsz), 256, 0, stream>>>(
            entities, Wae, bae, f_h, NE, f_sc);
        rowmax_kernel<<<Bsz, 256, 0, stream>>>(f_sc, NE, f_rmax);
        attend_part_kernel<<<dim3(NE / 128, Bsz), 256, 0, stream>>>(
            entities, f_sc, f_rmax, NE, NE / 128, f_part, f_psum);
        attend_reduce_kernel<<<dim3(Dm / 256, Bsz), 256, 0, stream>>>(
            f_part, f_psum, NE / 128, f_ectx);

        // ---- attend over sents -> sctx ----
        scores_kernel<<<dim3(NS / 64, Bsz), 256, 0, stream>>>(
            sents, Was, bas, f_h, NS, f_sc);
        rowmax_kernel<<<Bsz, 256, 0, stream>>>(f_sc, NS, f_rmax);
        attend_part_kernel<<<dim3(NS / 128, Bsz), 256, 0, stream>>>(
            sents, f_sc, f_rmax, NS, NS / 128, f_part, f_psum);
        attend_reduce_kernel<<<dim3(Dm / 256, Bsz), 256, 0, stream>>>(
            f_part, f_psum, NS / 128, f_sctx);

        // ---- gate + selection ----
        p_select_kernel<<<1, 256, 0, stream>>>(f_h, f_ectx, f_sctx, Wp, bp, i_top);
        gather_kernel<<<64, 256, 0, stream>>>(
            sents, f_ectx, f_h, i_top, t, f_xpad, f_hpad, out);

        // ---- LSTM step (WMMA fp32 GEMM + pointwise) ----
        gates_wmma_kernel<<<64, 128, 0, stream>>>(
            f_xpad, f_hpad, W_ih, W_hh, b_ih, b_hh, f_gate);
        lstm_kernel<<<32, 256, 0, stream>>>(f_gate, f_h, f_c);
    }
}